// ContextNorm_62783831933726
// MI455X (gfx1250) — compile-verified
//
#include <hip/hip_runtime.h>

typedef __attribute__((ext_vector_type(2))) float v2f;
typedef __attribute__((ext_vector_type(4))) float v4f;
typedef __attribute__((ext_vector_type(8))) float v8f;

#define KCTX 16
#define EPSV 1e-3f

// ---------------------------------------------------------------- zero ws
__global__ void ContextNorm_zero_kernel(float* __restrict__ ws, int n) {
    int i = blockIdx.x * blockDim.x + threadIdx.x;
    if (i < n) ws[i] = 0.0f;
}

// ---------------------------------------------------------------- stats
// Segment sums as one-hot GEMM on the matrix unit:
//   D(16 ctx x 16 chan) += A(16x4 one-hot of contexts) x B(4 samples x 16 chan)
// via V_WMMA_F32_16X16X4_F32.  Each wave owns one 16-channel slab; a block of
// 8 waves covers 128 channels; cgroups blocks cover all C channels of a chunk.
// Main loop is guard-free (chunks are 4-aligned); one guarded tail group.
__global__ __launch_bounds__(256) void ContextNorm_stats_kernel(
    const float* __restrict__ samples, const int* __restrict__ contexts,
    float* __restrict__ gS1, float* __restrict__ gS2, float* __restrict__ gCnt,
    int N, int C, int chunkSize, int cgroups) {
    const int chunk = blockIdx.x / cgroups;
    const int cg    = blockIdx.x % cgroups;
    const int wave  = threadIdx.x >> 5;
    const int lane  = threadIdx.x & 31;
    const int m     = lane & 15;   // A: row M; B/D: column N
    const int half  = lane >> 4;   // K-half for A/B, M-half for D
    const int cbase = cg * 128 + wave * 16;
    const int sBeg  = chunk * chunkSize;
    const int sEnd  = (sBeg + chunkSize < N) ? (sBeg + chunkSize) : N;
    const int span  = sEnd - sBeg;
    const int nGroups = span >> 2;       // full 4-sample groups
    const int rem     = span & 3;        // tail samples

    v8f acc1 = {};   // sum(x)
    v8f acc2 = {};   // sum(x*x)

    // Per-lane streaming pointers: lane's rows are (s + 2*half) and (s + 2*half + 1).
    const size_t rowStride = (size_t)C;
    const float* pB = samples + (size_t)(sBeg + 2 * half) * C + cbase + m;
    const int*   pC = contexts + sBeg + 2 * half;

#pragma unroll 4
    for (int g = 0; g < nGroups; ++g) {
        const int   c0 = pC[0];
        const int   c1 = pC[1];
        // samples stream is read exactly once device-wide -> non-temporal
        const float bx = __builtin_nontemporal_load(pB);
        const float by = __builtin_nontemporal_load(pB + rowStride);
        __builtin_prefetch(pB + 16 * rowStride, 0, 3);   // ~4 groups ahead, near-cache
        // A-layout (16x4 f32): VGPR0 holds K = 2*half, VGPR1 holds K = 2*half+1
        // B-layout (4x16 f32): same K striping; lane's column N = m
        v2f a = {(c0 == m) ? 1.0f : 0.0f, (c1 == m) ? 1.0f : 0.0f};
        v2f b = {bx, by};
        acc1 = __builtin_amdgcn_wmma_f32_16x16x4_f32(
            false, a, false, b, (short)0, acc1, false, false);
        v2f b2 = {bx * bx, by * by};
        acc2 = __builtin_amdgcn_wmma_f32_16x16x4_f32(
            false, a, false, b2, (short)0, acc2, false, false);
        pB += 4 * rowStride;
        pC += 4;
    }

    if (rem) {  // one guarded tail group (only when span % 4 != 0)
        const int sa = sBeg + 4 * nGroups + 2 * half;
        const int sb = sa + 1;
        v2f a = {0.0f, 0.0f};
        v2f b = {0.0f, 0.0f};
        if (sa < sEnd) {
            a.x = (contexts[sa] == m) ? 1.0f : 0.0f;
            b.x = samples[(size_t)sa * C + cbase + m];
        }
        if (sb < sEnd) {
            a.y = (contexts[sb] == m) ? 1.0f : 0.0f;
            b.y = samples[(size_t)sb * C + cbase + m];
        }
        acc1 = __builtin_amdgcn_wmma_f32_16x16x4_f32(
            false, a, false, b, (short)0, acc1, false, false);
        v2f b2 = {b.x * b.x, b.y * b.y};
        acc2 = __builtin_amdgcn_wmma_f32_16x16x4_f32(
            false, a, false, b2, (short)0, acc2, false, false);
    }

    // D-layout: VGPR v -> (M = v + 8*half, N = m)
    const int ch = cbase + m;
#pragma unroll
    for (int v = 0; v < 8; ++v) {
        const int k = v + 8 * half;
        unsafeAtomicAdd(&gS1[k * C + ch], acc1[v]);
        unsafeAtomicAdd(&gS2[k * C + ch], acc2[v]);
    }

    // Counts: only channel-group-0 blocks count (avoid cgroups-fold duplication)
    __shared__ int bins[KCTX];
    if (cg == 0) {
        if (threadIdx.x < KCTX) bins[threadIdx.x] = 0;
        __syncthreads();
        for (int s = sBeg + threadIdx.x; s < sEnd; s += blockDim.x)
            atomicAdd(&bins[contexts[s]], 1);
        __syncthreads();
        if (threadIdx.x < KCTX)
            unsafeAtomicAdd(&gCnt[threadIdx.x], (float)bins[threadIdx.x]);
    }
}

// ---------------------------------------------------------------- finalize
// Absent contexts fold to identity (w=1, b=0) so apply is branch-free.
__global__ void ContextNorm_finalize_kernel(
    const float* __restrict__ gS1, const float* __restrict__ gS2,
    const float* __restrict__ gCnt,
    const float* __restrict__ gamma, const float* __restrict__ beta,
    const float* __restrict__ priors,
    float* __restrict__ gW, float* __restrict__ gB, int C, int K) {
    int i = blockIdx.x * blockDim.x + threadIdx.x;
    if (i >= K * C) return;
    int k = i / C;
    float n       = gCnt[k];
    float safe_n  = fmaxf(n, 1.0f);
    float mean    = gS1[i] / safe_n;
    float var     = gS2[i] / safe_n - mean * mean;
    float inv_std = rsqrtf(var + EPSV);
    float ps      = rsqrtf(priors[k]);
    float w = gamma[i] * inv_std * ps;
    float b = (beta[i] - gamma[i] * mean * inv_std) * ps;
    if (n <= 0.0f) { w = 1.0f; b = 0.0f; }
    gW[i] = w;
    gB[i] = b;
}

// ---------------------------------------------------------------- apply
// out = samples * w[ctx] + b[ctx], b128 vectorized via clang ext-vectors.
// samples/out are touched once -> non-temporal; w/b are hot in L2/WGP$.
// c4n = C/4 is a power of two -> shift/mask addressing, no integer division.
__global__ __launch_bounds__(256) void ContextNorm_apply_kernel(
    const float* __restrict__ samples, const int* __restrict__ contexts,
    const float* __restrict__ gW, const float* __restrict__ gB,
    float* __restrict__ out, int N, int c4n, int c4shift, int rowsPerBlk) {
    const int row = blockIdx.x * rowsPerBlk + ((int)threadIdx.x >> c4shift);
    const int c4  = (int)threadIdx.x & (c4n - 1);
    if (row >= N) return;
    const int k = contexts[row];
    const v4f* s4 = (const v4f*)samples;
    const v4f* w4 = (const v4f*)gW;
    const v4f* b4 = (const v4f*)gB;
    v4f* o4 = (v4f*)out;
    const size_t idx = (size_t)row * c4n + c4;
    v4f s = __builtin_nontemporal_load(&s4[idx]);
    v4f w = w4[k * c4n + c4];
    v4f b = b4[k * c4n + c4];
    v4f o;
    o.x = fmaf(s.x, w.x, b.x);
    o.y = fmaf(s.y, w.y, b.y);
    o.z = fmaf(s.z, w.z, b.z);
    o.w = fmaf(s.w, w.w, b.w);
    __builtin_nontemporal_store(o, &o4[idx]);
}

// ---------------------------------------------------------------- launch
extern "C" void kernel_launch(void* const* d_in, const int* in_sizes, int n_in,
                              void* d_out, int out_size, void* d_ws, size_t ws_size,
                              hipStream_t stream) {
    const float* samples  = (const float*)d_in[0];
    const int*   contexts = (const int*)d_in[1];
    const float* gamma    = (const float*)d_in[2];
    const float* beta     = (const float*)d_in[3];
    const float* priors   = (const float*)d_in[4];
    float* out = (float*)d_out;

    const int N = in_sizes[1];
    const int K = in_sizes[4];
    const int C = in_sizes[2] / K;

    float* ws   = (float*)d_ws;
    float* gS1  = ws;                      // [K,C]
    float* gS2  = gS1 + (size_t)K * C;     // [K,C]
    float* gCnt = gS2 + (size_t)K * C;     // [K]
    float* gW   = gCnt + K;                // [K,C]
    float* gB   = gW + (size_t)K * C;      // [K,C]

    // 1) zero accumulators (ws is poisoned by the harness)
    const int nZero = 2 * K * C + K;
    ContextNorm_zero_kernel<<<(nZero + 255) / 256, 256, 0, stream>>>(ws, nZero);

    // 2) WMMA one-hot segment sums
    const int chunkSize = 1024;
    const int cgroups   = C / 128;  // 8 waves x 16 channels per block
    const int nChunks   = (N + chunkSize - 1) / chunkSize;
    ContextNorm_stats_kernel<<<nChunks * cgroups, 256, 0, stream>>>(
        samples, contexts, gS1, gS2, gCnt, N, C, chunkSize, cgroups);

    // 3) finalize w/b
    ContextNorm_finalize_kernel<<<(K * C + 255) / 256, 256, 0, stream>>>(
        gS1, gS2, gCnt, gamma, beta, priors, gW, gB, C, K);

    // 4) elementwise apply (C/4 is a power of two for this problem)
    const int c4n = C >> 2;
    int c4shift = 0;
    while ((1 << c4shift) < c4n) ++c4shift;
    const int rowsPerBlk = (256 * 4) / C;
    ContextNorm_apply_kernel<<<(N + rowsPerBlk - 1) / rowsPerBlk, 256, 0, stream>>>(
        samples, contexts, gW, gB, out, N, c4n, c4shift, rowsPerBlk);
}